// TripletNN_38517266710626
// MI455X (gfx1250) — compile-verified
//
#include <hip/hip_runtime.h>
#include <hip/hip_bf16.h>

typedef __attribute__((ext_vector_type(2))) float v2f;
typedef __attribute__((ext_vector_type(4))) float v4f;
typedef __attribute__((ext_vector_type(8))) float v8f;

#define Bsz 4
#define Nn  128
#define Ff  128
#define Tt  8

// ---- workspace layout (floats) ----
#define OFF_TE1T 0u
#define OFF_TE2T 524288u            // 4*128*128*8
#define OFF_TE3  1048576u
#define OFF_TZ1  1572864u
#define OFF_TZ2  1576960u           // +4096
#define OFF_TZ3  1581056u
#define OFF_TG   1585152u
#define OFF_WE1P 1585184u           // 16x128 zero-padded We copies
#define OFF_WE2P 1587232u
#define OFF_WE3P 1589280u

// -----------------------------------------------------------------------------
// Setup: zero-padded 16x128 copies of We1/2/3 (rows 8..15 = 0) so the WMMA
// B-operand loads need no per-step masking.
// -----------------------------------------------------------------------------
__global__ __launch_bounds__(256)
void k_pad_weights(const float* __restrict__ We1, const float* __restrict__ We2,
                   const float* __restrict__ We3,
                   float* __restrict__ We1p, float* __restrict__ We2p,
                   float* __restrict__ We3p) {
    const int idx = blockIdx.x * 256 + threadIdx.x;  // 0..2047
    const int m = idx >> 7;
    const int k = idx & 127;
    const bool live = (m < Tt);
    We1p[idx] = live ? We1[m * Ff + k] : 0.0f;
    We2p[idx] = live ? We2[m * Ff + k] : 0.0f;
    We3p[idx] = live ? We3[m * Ff + k] : 0.0f;
}

// -----------------------------------------------------------------------------
// Kernel 1: edge projections te1/te2/te3 = edge_feats @ We{1,2,3}^T + be
// Rows = flattened (b,x,y), 65536 rows of K=128. One wave per 16-row tile,
// V_WMMA_F32_16X16X4_F32 chained 32x over K, three accumulators (one per We).
// te1,te2 stored transposed [b][y][x][t]; te3 stored direct [b][x][y][t].
// -----------------------------------------------------------------------------
__global__ __launch_bounds__(256)
void k_edge_proj(const float* __restrict__ edge,
                 const float* __restrict__ We1p, const float* __restrict__ be1,
                 const float* __restrict__ We2p, const float* __restrict__ be2,
                 const float* __restrict__ We3p, const float* __restrict__ be3,
                 float* __restrict__ te1T, float* __restrict__ te2T,
                 float* __restrict__ te3) {
    const int wave = threadIdx.x >> 5;
    const int lane = threadIdx.x & 31;
    const int tile = blockIdx.x * 8 + wave;          // 4096 tiles of 16 rows
    const int rowBase = tile * 16;
    const int m  = lane & 15;                        // N-col of B/C, M-row of A
    const int hi = lane >> 4;                        // lane half -> K 0,1 vs 2,3

    const float* arow = edge + (size_t)(rowBase + m) * Ff + hi * 2;
    const float* wp1  = We1p + m * Ff + hi * 2;      // rows 8..15 pre-zeroed
    const float* wp2  = We2p + m * Ff + hi * 2;
    const float* wp3  = We3p + m * Ff + hi * 2;

    v8f c1 = {}, c2 = {}, c3 = {};
    #pragma unroll 4
    for (int k0 = 0; k0 < Ff; k0 += 4) {
        v2f a  = *(const v2f*)(arow + k0);
        v2f b1 = *(const v2f*)(wp1 + k0);
        v2f b2 = *(const v2f*)(wp2 + k0);
        v2f b3 = *(const v2f*)(wp3 + k0);
        c1 = __builtin_amdgcn_wmma_f32_16x16x4_f32(false, a, false, b1, (short)0, c1, false, false);
        c2 = __builtin_amdgcn_wmma_f32_16x16x4_f32(false, a, false, b2, (short)0, c2, false, false);
        c3 = __builtin_amdgcn_wmma_f32_16x16x4_f32(false, a, false, b3, (short)0, c3, false, false);
    }

    if (m < Tt) {
        const float o1 = be1[m], o2 = be2[m], o3 = be3[m];
        #pragma unroll
        for (int r = 0; r < 8; ++r) {
            const int row = rowBase + r + hi * 8;    // C/D: M = r + 8*hi
            const int b   = row >> 14;
            const int x   = (row >> 7) & 127;
            const int y   = row & 127;
            te1T[(((b * Nn + y) * Nn + x) << 3) + m] = c1[r] + o1;
            te2T[(((b * Nn + y) * Nn + x) << 3) + m] = c2[r] + o2;
            te3 [((size_t)row << 3) + m]             = c3[r] + o3;
        }
    }
}

// -----------------------------------------------------------------------------
// Kernel 2: node projections tz1/2/3 (B*N rows, tiny) -- plain VALU dot products
// -----------------------------------------------------------------------------
__global__ __launch_bounds__(256)
void k_z_proj(const float* __restrict__ z,
              const float* __restrict__ Wz1, const float* __restrict__ bz1,
              const float* __restrict__ Wz2, const float* __restrict__ bz2,
              const float* __restrict__ Wz3, const float* __restrict__ bz3,
              float* __restrict__ tz1, float* __restrict__ tz2,
              float* __restrict__ tz3) {
    const int id  = blockIdx.x * 256 + threadIdx.x;  // 4096 = 512 rows * 8 t
    const int row = id >> 3;
    const int t   = id & 7;
    const float* zr = z + (size_t)row * Ff;
    const float* w1 = Wz1 + t * Ff;
    const float* w2 = Wz2 + t * Ff;
    const float* w3 = Wz3 + t * Ff;
    float a1 = 0.f, a2 = 0.f, a3 = 0.f;
    #pragma unroll 8
    for (int k = 0; k < Ff; k += 4) {
        v4f zv = *(const v4f*)(zr + k);
        v4f v1 = *(const v4f*)(w1 + k);
        v4f v2 = *(const v4f*)(w2 + k);
        v4f v3 = *(const v4f*)(w3 + k);
        a1 += zv.x*v1.x + zv.y*v1.y + zv.z*v1.z + zv.w*v1.w;
        a2 += zv.x*v2.x + zv.y*v2.y + zv.z*v2.z + zv.w*v2.w;
        a3 += zv.x*v3.x + zv.y*v3.y + zv.z*v3.z + zv.w*v3.w;
    }
    tz1[id] = a1 + bz1[t];
    tz2[id] = a2 + bz2[t];
    tz3[id] = a3 + bz3[t];
}

// graph projection tg = graph_feats @ Wg^T + bg : 32 values total
__global__ void k_g_proj(const float* __restrict__ g,
                         const float* __restrict__ Wg,
                         const float* __restrict__ bg,
                         float* __restrict__ tg) {
    const int id = threadIdx.x;                      // 32 threads
    const int b  = id >> 3;
    const int t  = id & 7;
    const float* gr = g + b * Ff;
    const float* wr = Wg + t * Ff;
    float a = 0.f;
    for (int k = 0; k < Ff; ++k) a += gr[k] * wr[k];
    tg[id] = a + bg[t];
}

// -----------------------------------------------------------------------------
// Kernel 3: fused max-over-i + epilogue + output linear (WMMA) + ReLU.
// One block per (b,j). Phase 1: LDS s_a[i][t] = tz1[b,i,t] + te1T[b,j,i,t].
// Phase 2: per (k, t-quad) thread: max_i(s_a + te2T[b,k,i,t]) then add
//          tz2/tz3/te3/tg -> s_s[k][t]. Phase 3: (128x8)@(8x128) via
//          V_WMMA_F32_16X16X4_F32, one M-tile per wave, +bo, ReLU, store.
// -----------------------------------------------------------------------------
__global__ __launch_bounds__(256)
void k_main(const float* __restrict__ te1T, const float* __restrict__ te2T,
            const float* __restrict__ te3,
            const float* __restrict__ tz1, const float* __restrict__ tz2,
            const float* __restrict__ tz3, const float* __restrict__ tg,
            const float* __restrict__ Wo,  const float* __restrict__ bo,
            float* __restrict__ out) {
    __shared__ float s_a[Nn * Tt];   // 4 KB
    __shared__ float s_s[Nn * Tt];   // 4 KB

    const int tid = threadIdx.x;
    const int bj  = blockIdx.x;                      // 512 = B * N
    const int b   = bj >> 7;
    const int j   = bj & 127;

    // Phase 1: stage tz1[b] + te1T[b][j] (1024 floats) into LDS
    {
        const float* p1 = te1T + ((size_t)bj << 10); // (b*N + j) * N * 8
        const float* p2 = tz1  + (b << 10);
        v4f v = *(const v4f*)(p1 + tid * 4) + *(const v4f*)(p2 + tid * 4);
        *(v4f*)(&s_a[tid * 4]) = v;
    }
    __syncthreads();

    // Phase 2: max over i, then epilogue adds, into s_s
    {
        const int k    = tid >> 1;
        const int tgrp = (tid & 1) * 4;
        const float* e2 = te2T + (((size_t)(b * Nn + k)) << 10) + tgrp;
        float m0 = -3.402823466e38f, m1 = m0, m2 = m0, m3 = m0;
        #pragma unroll 4
        for (int i = 0; i < Nn; ++i) {
            v4f e = *(const v4f*)(e2 + i * Tt);
            v4f a = *(const v4f*)(&s_a[i * Tt + tgrp]);
            m0 = fmaxf(m0, a.x + e.x);
            m1 = fmaxf(m1, a.y + e.y);
            m2 = fmaxf(m2, a.z + e.z);
            m3 = fmaxf(m3, a.w + e.w);
        }
        const float* z2 = tz2 + (bj << 3) + tgrp;                        // [b,j,t]
        const float* z3 = tz3 + (((b << 7) + k) << 3) + tgrp;            // [b,k,t]
        const float* e3 = te3 + (((size_t)bj * Nn + k) << 3) + tgrp;     // [b,j,k,t]
        const float* gg = tg + (b << 3) + tgrp;
        s_s[k * Tt + tgrp + 0] = m0 + z2[0] + z3[0] + e3[0] + gg[0];
        s_s[k * Tt + tgrp + 1] = m1 + z2[1] + z3[1] + e3[1] + gg[1];
        s_s[k * Tt + tgrp + 2] = m2 + z2[2] + z3[2] + e3[2] + gg[2];
        s_s[k * Tt + tgrp + 3] = m3 + z2[3] + z3[3] + e3[3] + gg[3];
    }
    __syncthreads();

    // Phase 3: out[b,j,k,:] = relu(s_s @ Wo^T + bo) with f32 WMMA.
    {
        const int wave = tid >> 5;                   // M-tile (k rows 16w..16w+15)
        const int lane = tid & 31;
        const int m    = lane & 15;
        const int hi   = lane >> 4;
        float* outBJ = out + ((size_t)bj << 14);     // (b*N + j) * N * 128
        #pragma unroll
        for (int ft = 0; ft < 8; ++ft) {
            const int f = ft * 16 + m;
            v8f c = {};
            #pragma unroll
            for (int ks = 0; ks < 2; ++ks) {
                v2f a  = *(const v2f*)(&s_s[(wave * 16 + m) * Tt + ks * 4 + hi * 2]);
                v2f bb = *(const v2f*)(Wo + f * Tt + ks * 4 + hi * 2);
                c = __builtin_amdgcn_wmma_f32_16x16x4_f32(false, a, false, bb, (short)0, c, false, false);
            }
            const float bias = bo[f];
            #pragma unroll
            for (int r = 0; r < 8; ++r) {
                const int kk = wave * 16 + r + hi * 8;
                const float v = c[r] + bias;
                outBJ[kk * Ff + f] = v > 0.f ? v : 0.f;
            }
        }
    }
}

extern "C" void kernel_launch(void* const* d_in, const int* in_sizes, int n_in,
                              void* d_out, int out_size, void* d_ws, size_t ws_size,
                              hipStream_t stream) {
    const float* z    = (const float*)d_in[0];
    const float* edge = (const float*)d_in[1];
    const float* gf   = (const float*)d_in[2];
    const float* Wz1  = (const float*)d_in[3];
    const float* bz1  = (const float*)d_in[4];
    const float* Wz2  = (const float*)d_in[5];
    const float* bz2  = (const float*)d_in[6];
    const float* Wz3  = (const float*)d_in[7];
    const float* bz3  = (const float*)d_in[8];
    const float* We1  = (const float*)d_in[9];
    const float* be1  = (const float*)d_in[10];
    const float* We2  = (const float*)d_in[11];
    const float* be2  = (const float*)d_in[12];
    const float* We3  = (const float*)d_in[13];
    const float* be3  = (const float*)d_in[14];
    const float* Wg   = (const float*)d_in[15];
    const float* bg   = (const float*)d_in[16];
    const float* Wo   = (const float*)d_in[17];
    const float* bo   = (const float*)d_in[18];
    float* out = (float*)d_out;
    float* ws  = (float*)d_ws;

    float* te1T = ws + OFF_TE1T;
    float* te2T = ws + OFF_TE2T;
    float* te3  = ws + OFF_TE3;
    float* tz1  = ws + OFF_TZ1;
    float* tz2  = ws + OFF_TZ2;
    float* tz3  = ws + OFF_TZ3;
    float* tg   = ws + OFF_TG;
    float* We1p = ws + OFF_WE1P;
    float* We2p = ws + OFF_WE2P;
    float* We3p = ws + OFF_WE3P;

    k_pad_weights<<<8, 256, 0, stream>>>(We1, We2, We3, We1p, We2p, We3p);
    // 65536 edge rows / 16 per wave / 8 waves per block = 512 blocks
    k_edge_proj<<<512, 256, 0, stream>>>(edge, We1p, be1, We2p, be2, We3p, be3,
                                         te1T, te2T, te3);
    k_z_proj<<<16, 256, 0, stream>>>(z, Wz1, bz1, Wz2, bz2, Wz3, bz3,
                                     tz1, tz2, tz3);
    k_g_proj<<<1, 32, 0, stream>>>(gf, Wg, bg, tg);
    k_main<<<Bsz * Nn, 256, 0, stream>>>(te1T, te2T, te3, tz1, tz2, tz3, tg,
                                         Wo, bo, out);
}